// Decoder_24146306138719
// MI455X (gfx1250) — compile-verified
//
#include <hip/hip_runtime.h>
#include <math.h>

// Problem constants (match reference)
#define N     1024
#define NMAT  16
#define TOPK  8
#define PB    16          // LU panel width (= WMMA K)

// GEMM tiling
#define KC    16
#define BM    128
#define BN    64
#define LDSS  (KC + 2)    // 18-float row stride: even (float2-aligned), conflict-spread
#define NCH   (6 * (N / KC))   // 6 pairs x 64 K-chunks

typedef float v2f __attribute__((ext_vector_type(2)));
typedef float v8f __attribute__((ext_vector_type(8)));

// ---------------------------------------------------------------------------
// Kernel 1: blocked LU (panel=16) with partial pivoting -> log|det|.
// One block (32 waves) per matrix. Panel: shfl-based argmax + restricted
// rank-1 updates. U12: register triangular solve (no inner barriers).
// Trailing A22 -= L21*U12: WMMA rank-16 update (A-frag negated at load).
// ---------------------------------------------------------------------------
__global__ __launch_bounds__(1024)
void lu_logdet_kernel(const float* __restrict__ x, float* __restrict__ lu,
                      float* __restrict__ logdet) {
  const int t = threadIdx.x;
  const int nt = 1024;
  const int lane = t & 31;
  const int wave = t >> 5;                 // 32 waves
  float* A = lu + (size_t)blockIdx.x * N * N;
  const float* X = x + (size_t)blockIdx.x * N * N;
  for (int i = t; i < N * N; i += nt) A[i] = X[i];
  __syncthreads();

  __shared__ float s_red[32];
  __shared__ int   s_redi[32];
  __shared__ float s_prow[PB];
  __shared__ float s_diag;
  __shared__ int   s_pr;
  __shared__ float s_L11[PB][PB];

  float acc = 0.0f;

  for (int pb = 0; pb < N; pb += PB) {
    // ---------------- panel factorization (columns pb .. pb+15) ------------
    for (int kk = 0; kk < PB; ++kk) {
      const int k = pb + kk;
      // argmax |A[i][k]|, i in [k, N): thread-local, then shfl wave-reduce
      float best = -1.0f; int bi = k;
      for (int i = k + t; i < N; i += nt) {
        float v = fabsf(A[(size_t)i * N + k]);
        if (v > best) { best = v; bi = i; }
      }
      for (int off = 16; off > 0; off >>= 1) {
        float ov = __shfl_xor(best, off, 32);
        int   oi = __shfl_xor(bi,   off, 32);
        if (ov > best || (ov == best && oi < bi)) { best = ov; bi = oi; }
      }
      if (lane == 0) { s_red[wave] = best; s_redi[wave] = bi; }
      __syncthreads();
      if (t < 32) {                        // wave 0 reduces the 32 partials
        best = s_red[t]; bi = s_redi[t];
        for (int off = 16; off > 0; off >>= 1) {
          float ov = __shfl_xor(best, off, 32);
          int   oi = __shfl_xor(bi,   off, 32);
          if (ov > best || (ov == best && oi < bi)) { best = ov; bi = oi; }
        }
        if (t == 0) s_pr = bi;
      }
      __syncthreads();
      const int pr = s_pr;

      if (pr != k) {                       // full-row swap
        for (int j = t; j < N; j += nt) {
          float tmp = A[(size_t)k * N + j];
          A[(size_t)k * N + j]  = A[(size_t)pr * N + j];
          A[(size_t)pr * N + j] = tmp;
        }
      }
      __syncthreads();

      if (t == 0) s_diag = A[(size_t)k * N + k];
      if (t < PB - 1 - kk) s_prow[t] = A[(size_t)k * N + (k + 1) + t];
      __syncthreads();

      const float diag = s_diag;
      if (t == 0) acc += logf(fabsf(diag));
      const float inv = 1.0f / diag;
      // multipliers + rank-1 update restricted to the panel columns
      const int jend = PB - 1 - kk;
      for (int i = k + 1 + t; i < N; i += nt) {
        float m = A[(size_t)i * N + k] * inv;
        A[(size_t)i * N + k] = m;
        float* Ai = A + (size_t)i * N + (k + 1);
        for (int j = 0; j < jend; ++j) Ai[j] -= m * s_prow[j];
      }
      __syncthreads();
    }

    const int r0 = pb + PB;
    if (r0 >= N) break;                    // last panel: no trailing part
    const int nrem = N - r0;

    // ---------------- stage L11, solve U12 = L11^{-1} * A12 ----------------
    if (t < PB * PB) {
      int r = t / PB, c = t % PB;
      s_L11[r][c] = A[(size_t)(pb + r) * N + pb + c];
    }
    __syncthreads();

    for (int j = r0 + t; j < N; j += nt) { // one column per thread, in regs
      float u[PB];
      for (int m = 0; m < PB; ++m) {
        float val = A[(size_t)(pb + m) * N + j];
        for (int q = 0; q < m; ++q) val -= s_L11[m][q] * u[q];
        u[m] = val;
      }
      for (int m = 0; m < PB; ++m) A[(size_t)(pb + m) * N + j] = u[m];
    }
    __syncthreads();

    // ---------------- trailing update: A22 -= L21 * U12 (WMMA, K=16) -------
    const int ntr = nrem / 16, ntc = nrem / 16;
    const int lm = lane & 15, kh = lane >> 4;
    for (int tr = wave; tr < ntr; tr += 32) {
      const int rb = r0 + tr * 16;
      v2f af[4];                           // -L21 frags, reused across the row
      for (int ks = 0; ks < 4; ++ks) {
        const int k0 = ks * 4 + kh * 2;
        af[ks][0] = -A[(size_t)(rb + lm) * N + pb + k0];
        af[ks][1] = -A[(size_t)(rb + lm) * N + pb + k0 + 1];
      }
      for (int tc = 0; tc < ntc; ++tc) {
        const int cb = r0 + tc * 16;
        v8f cacc;
        for (int r = 0; r < 8; ++r)
          cacc[r] = A[(size_t)(rb + r + kh * 8) * N + cb + lm];
        for (int ks = 0; ks < 4; ++ks) {
          const int k0 = ks * 4 + kh * 2;
          v2f bf;
          bf[0] = A[(size_t)(pb + k0) * N + cb + lm];
          bf[1] = A[(size_t)(pb + k0 + 1) * N + cb + lm];
          cacc = __builtin_amdgcn_wmma_f32_16x16x4_f32(
              false, af[ks], false, bf, (short)0, cacc, false, false);
        }
        for (int r = 0; r < 8; ++r)
          A[(size_t)(rb + r + kh * 8) * N + cb + lm] = cacc[r];
      }
    }
    __syncthreads();
  }
  if (t == 0) logdet[blockIdx.x] = acc;
}

// ---------------------------------------------------------------------------
// Kernel 2: gate, top-8 selection, v = w2@w1, const = w2@b1 + b2, flag output.
// ---------------------------------------------------------------------------
__global__ void prep_kernel(const unsigned char* __restrict__ flags,
                            const float* __restrict__ logdet,
                            const float* __restrict__ w1,   // (32,10)
                            const float* __restrict__ b1,   // (32)
                            const float* __restrict__ w2,   // (32)
                            const float* __restrict__ b2,   // (1)
                            int* __restrict__ meta,         // [0]=gate, [1..8]=idx
                            float* __restrict__ vscale,     // [0..9]=v, [10]=const
                            float* __restrict__ out) {
  if (threadIdx.x != 0 || blockIdx.x != 0) return;
  int cnt = 0;
  for (int i = 0; i < NMAT; ++i) cnt += flags[i] ? 1 : 0;
  const int gate = (cnt >= 4) ? 1 : 0;

  float scores[NMAT];
  for (int i = 0; i < NMAT; ++i)
    scores[i] = flags[i] ? logdet[i] : -INFINITY;

  bool used[NMAT];
  for (int i = 0; i < NMAT; ++i) used[i] = false;
  for (int r = 0; r < TOPK; ++r) {
    int bi = -1; float best = 0.0f;
    for (int i = 0; i < NMAT; ++i) {
      if (!used[i] && (bi < 0 || scores[i] > best)) { bi = i; best = scores[i]; }
    }
    used[bi] = true;
    meta[1 + r] = bi;
  }
  meta[0] = gate;

  for (int c = 0; c < 10; ++c) {
    float s = 0.0f;
    for (int h = 0; h < 32; ++h) s += w2[h] * w1[h * 10 + c];
    vscale[c] = s;
  }
  float cst = b2[0];
  for (int h = 0; h < 32; ++h) cst += w2[h] * b1[h];
  vscale[10] = cst;

  out[(size_t)N * N] = gate ? 1.0f : 0.0f;  // out_active
}

// ---------------------------------------------------------------------------
// Kernel 3: fused 6-pair fp32 WMMA GEMM, double-buffered:
//   A tile  : async global->LDS (global_load_async_to_lds_b128, ASYNCcnt)
//   B tile  : register prefetch -> scaled, transposed LDS store
//   out = const + sum_p v[p]*(A_p x B_p) + sum_j v[6+j]*preserve_j
// ---------------------------------------------------------------------------
__global__ __launch_bounds__(256)
void fused_pair_gemm_kernel(const float* __restrict__ x,
                            const int* __restrict__ meta,
                            const float* __restrict__ vscale,
                            float* __restrict__ out) {
  const int t = threadIdx.x;
  const int blockM = blockIdx.y * BM;
  const int blockN = blockIdx.x * BN;

  const int gate = meta[0];
  if (!gate) {                              // inactive path -> zeros
    for (int e = t; e < BM * BN; e += 256)
      out[(size_t)(blockM + e / BN) * N + blockN + (e % BN)] = 0.0f;
    return;
  }

  __shared__ __align__(16) float As[2][BM * LDSS];
  __shared__ __align__(16) float Bt[2][BN * LDSS];  // transposed [n][k]

  int idx[TOPK];
  for (int i = 0; i < TOPK; ++i) idx[i] = meta[1 + i];
  float v[10];
  for (int i = 0; i < 10; ++i) v[i] = vscale[i];
  const float cst = vscale[10];

  const int pa[6] = {0, 0, 0, 1, 1, 2};     // triu_indices(4, k=1)
  const int pb[6] = {1, 2, 3, 2, 3, 3};

  const int lane = t & 31;
  const int wave = t >> 5;
  const int wm0  = (wave & 3) * 32;
  const int wn0  = (wave >> 2) * 32;
  const int lm   = lane & 15;
  const int kh   = lane >> 4;

  // B prefetch coords: one float4 per thread per chunk
  const int bRow  = t / (BN / 4);           // k row 0..15
  const int bCol  = (t % (BN / 4)) * 4;     // n col

  const v8f vzero = {0.f, 0.f, 0.f, 0.f, 0.f, 0.f, 0.f, 0.f};
  v8f acc[2][2];
  acc[0][0] = vzero; acc[0][1] = vzero; acc[1][0] = vzero; acc[1][1] = vzero;

  // ---- prologue: stage chunk 0 ----
  {
    const float* Ap = x + (size_t)idx[pa[0]] * N * N;
    const float* Bp = x + (size_t)idx[pb[0]] * N * N;
    const float s = v[0];
    for (int e = t; e < (BM * KC) / 4; e += 256) {
      const int row = e >> 2, col4 = (e & 3) * 4;
      const float* src = Ap + (size_t)(blockM + row) * N + col4;
      unsigned dst = (unsigned)(size_t)(void*)&As[0][row * LDSS + col4];
      asm volatile("global_load_async_to_lds_b128 %0, %1, off"
                   :: "v"(dst), "v"(src) : "memory");
    }
    const float4 b4 = *(const float4*)(Bp + (size_t)bRow * N + blockN + bCol);
    Bt[0][(bCol + 0) * LDSS + bRow] = b4.x * s;
    Bt[0][(bCol + 1) * LDSS + bRow] = b4.y * s;
    Bt[0][(bCol + 2) * LDSS + bRow] = b4.z * s;
    Bt[0][(bCol + 3) * LDSS + bRow] = b4.w * s;
  }

  for (int cc = 0; cc < NCH; ++cc) {
    const int buf = cc & 1;
    asm volatile("s_wait_asynccnt 0" ::: "memory");
    __syncthreads();                        // As[buf], Bt[buf] ready everywhere

    float4 b4;
    float sNext = 0.0f;
    const bool more = (cc + 1) < NCH;
    if (more) {                             // stage chunk cc+1 into buf^1
      const int cc2 = cc + 1;
      const int p2 = cc2 >> 6;
      const int kc2 = (cc2 & 63) * KC;
      const float* Ap = x + (size_t)idx[pa[p2]] * N * N;
      const float* Bp = x + (size_t)idx[pb[p2]] * N * N;
      sNext = v[p2];
      for (int e = t; e < (BM * KC) / 4; e += 256) {
        const int row = e >> 2, col4 = (e & 3) * 4;
        const float* src = Ap + (size_t)(blockM + row) * N + kc2 + col4;
        unsigned dst = (unsigned)(size_t)(void*)&As[buf ^ 1][row * LDSS + col4];
        asm volatile("global_load_async_to_lds_b128 %0, %1, off"
                     :: "v"(dst), "v"(src) : "memory");
      }
      b4 = *(const float4*)(Bp + (size_t)(kc2 + bRow) * N + blockN + bCol);
    }

    // ---- compute on As[buf], Bt[buf] ----
    for (int ks = 0; ks < KC / 4; ++ks) {
      const int k0 = ks * 4 + kh * 2;
      v2f afrag[2], bfrag[2];
      for (int mt = 0; mt < 2; ++mt)
        afrag[mt] = *(const v2f*)(&As[buf][(wm0 + mt * 16 + lm) * LDSS + k0]);
      for (int nt = 0; nt < 2; ++nt)
        bfrag[nt] = *(const v2f*)(&Bt[buf][(wn0 + nt * 16 + lm) * LDSS + k0]);
      for (int mt = 0; mt < 2; ++mt)
        for (int nt = 0; nt < 2; ++nt)
          acc[mt][nt] = __builtin_amdgcn_wmma_f32_16x16x4_f32(
              false, afrag[mt], false, bfrag[nt], (short)0, acc[mt][nt],
              false, false);
    }

    if (more) {                             // scaled, transposed B store
      Bt[buf ^ 1][(bCol + 0) * LDSS + bRow] = b4.x * sNext;
      Bt[buf ^ 1][(bCol + 1) * LDSS + bRow] = b4.y * sNext;
      Bt[buf ^ 1][(bCol + 2) * LDSS + bRow] = b4.z * sNext;
      Bt[buf ^ 1][(bCol + 3) * LDSS + bRow] = b4.w * sNext;
    }
  }

  // ---- epilogue: + const + scaled preserve matrices ----
  const float pv0 = v[6], pv1 = v[7], pv2 = v[8], pv3 = v[9];
  const float* P0 = x + (size_t)idx[4] * N * N;
  const float* P1 = x + (size_t)idx[5] * N * N;
  const float* P2 = x + (size_t)idx[6] * N * N;
  const float* P3 = x + (size_t)idx[7] * N * N;

  for (int mt = 0; mt < 2; ++mt) {
    for (int nt = 0; nt < 2; ++nt) {
      const int gn = blockN + wn0 + nt * 16 + lm;
      for (int r = 0; r < 8; ++r) {
        const int gm = blockM + wm0 + mt * 16 + r + kh * 8;
        const size_t off = (size_t)gm * N + gn;
        float val = acc[mt][nt][r] + cst;
        val += pv0 * P0[off] + pv1 * P1[off] + pv2 * P2[off] + pv3 * P3[off];
        out[off] = val;
      }
    }
  }
}

// ---------------------------------------------------------------------------
extern "C" void kernel_launch(void* const* d_in, const int* in_sizes, int n_in,
                              void* d_out, int out_size, void* d_ws, size_t ws_size,
                              hipStream_t stream) {
  const float*         x     = (const float*)d_in[0];          // (16,1024,1024)
  const unsigned char* flags = (const unsigned char*)d_in[1];  // (16,) bool
  const float*         w1    = (const float*)d_in[2];          // (32,10)
  const float*         b1    = (const float*)d_in[3];          // (32,1,1)
  const float*         w2    = (const float*)d_in[4];          // (1,32)
  const float*         b2    = (const float*)d_in[5];          // (1,1,1)
  float* out = (float*)d_out;                                  // N*N + 1 (flag)

  // Workspace layout: [LU scratch 64MB][logdet 16f][meta 16i][vscale 16f]
  char* ws = (char*)d_ws;
  float* lu     = (float*)ws;
  float* logdet = (float*)(ws + (size_t)NMAT * N * N * sizeof(float));
  int*   meta   = (int*)(logdet + 16);
  float* vscale = (float*)(meta + 16);

  lu_logdet_kernel<<<NMAT, 1024, 0, stream>>>(x, lu, logdet);
  prep_kernel<<<1, 1, 0, stream>>>(flags, logdet, w1, b1, w2, b2,
                                   meta, vscale, out);
  dim3 grid(N / BN, N / BM);  // 16 x 8 = 128 blocks
  fused_pair_gemm_kernel<<<grid, 256, 0, stream>>>(x, meta, vscale, out);
}